// KohonenSOM_25220047962466
// MI455X (gfx1250) — compile-verified
//
#include <hip/hip_runtime.h>

typedef __attribute__((ext_vector_type(2))) float v2f;
typedef __attribute__((ext_vector_type(4))) float v4f;
typedef __attribute__((ext_vector_type(8))) float v8f;

#define D_DIM 32

// ---------------------------------------------------------------------------
// Prologue: per-row squared L2 norm (rows are exactly 32 f32 = 128B).
// ---------------------------------------------------------------------------
__global__ void som_rownorm_kernel(const float* __restrict__ src,
                                   float* __restrict__ dst, int nrows) {
    int i = blockIdx.x * blockDim.x + threadIdx.x;
    if (i >= nrows) return;
    const v4f* p = (const v4f*)(src + (size_t)i * D_DIM);
    float s = 0.0f;
#pragma unroll
    for (int d = 0; d < D_DIM / 4; ++d) {
        v4f v = p[d];
        s += v.x * v.x + v.y * v.y + v.z * v.z + v.w * v.w;
    }
    dst[i] = s;
}

// ---------------------------------------------------------------------------
// Main kernel: cross = x . w^T via V_WMMA_F32_16X16X4_F32 (K=32 -> 8 steps),
// then D = sqrt(max(|x|^2 + |w|^2 - 2*cross, 0)).
//
// Fragment layouts (CDNA5 ISA 7.12.2, 32-bit operands, wave32):
//   A 16x4 (MxK):  lane lo = M row;  VGPR0/1 hold K = {2*hi, 2*hi+1}
//   B 4x16 (KxN):  lane lo = N col;  VGPR0/1 hold K = {2*hi, 2*hi+1}
//   C/D 16x16:     VGPR j, lane -> (M = j + 8*hi, N = lo)
// where hi = lane>=16, lo = lane&15.
// ---------------------------------------------------------------------------
__global__ __launch_bounds__(256) void som_dist_kernel(
    const float* __restrict__ x, const float* __restrict__ w,
    const float* __restrict__ x2, const float* __restrict__ w2,
    float* __restrict__ out, int Bdim, int Ndim, int ntiles_per_blk) {
    const int lane = threadIdx.x & 31;
    const int wave = threadIdx.x >> 5;
    const int lo   = lane & 15;
    const int hi   = lane >> 4;

    const int mtile = blockIdx.y * 8 + wave;   // one 16-row tile per wave
    const int mbase = mtile * 16;
    if (mbase >= Bdim) return;                 // wave-uniform: EXEC stays full

    // --- A fragments for all 8 K-steps (reused across every N tile) ---
    int arow = mbase + lo;
    if (arow >= Bdim) arow = Bdim - 1;
    const float* ap = x + (size_t)arow * D_DIM + 2 * hi;
    v2f a[8];
#pragma unroll
    for (int j = 0; j < 8; ++j) a[j] = *(const v2f*)(ap + 4 * j);

    // --- |x|^2 for this half-wave's 8 output rows (contiguous) ---
    const float* xsrc = x2 + mbase + 8 * hi;
    v4f xs0 = *(const v4f*)(xsrc);
    v4f xs1 = *(const v4f*)(xsrc + 4);
    float xsv[8] = {xs0.x, xs0.y, xs0.z, xs0.w, xs1.x, xs1.y, xs1.z, xs1.w};

    const int n_tiles = (Ndim + 15) / 16;
    for (int t = 0; t < ntiles_per_blk; ++t) {
        int ntile = blockIdx.x * ntiles_per_blk + t;
        if (ntile >= n_tiles) break;           // wave-uniform
        int nbase = ntile * 16;

        int  ncol  = nbase + lo;
        bool colok = (ncol < Ndim);
        int  ncl   = colok ? ncol : (Ndim - 1);  // clamp edge reads in-bounds

        const float* bp = w + (size_t)ncl * D_DIM + 2 * hi;
        v2f b[8];
#pragma unroll
        for (int j = 0; j < 8; ++j) b[j] = *(const v2f*)(bp + 4 * j);
        float w2v = w2[ncl];

        // Two accumulators for ILP across the serial K chain.
        v8f c0 = {};
        v8f c1 = {};
#pragma unroll
        for (int j = 0; j < 8; j += 2) {
            c0 = __builtin_amdgcn_wmma_f32_16x16x4_f32(
                false, a[j],     false, b[j],     (short)0, c0, false, false);
            c1 = __builtin_amdgcn_wmma_f32_16x16x4_f32(
                false, a[j + 1], false, b[j + 1], (short)0, c1, false, false);
        }

        if (colok) {
            float* op = out + (size_t)(mbase + 8 * hi) * Ndim + ncol;
#pragma unroll
            for (int j = 0; j < 8; ++j) {
                float cr = c0[j] + c1[j];
                float d2 = fmaf(-2.0f, cr, xsv[j] + w2v);
                d2 = fmaxf(d2, 0.0f);
                // Single v_sqrt_f32 (~1 ulp) instead of libm's correctly
                // rounded multi-instruction expansion: the epilogue must stay
                // under the HBM store roofline (~55us for 1.285 GB output).
                op[(size_t)j * Ndim] = __builtin_amdgcn_sqrtf(d2);
            }
        }
    }
}

// ---------------------------------------------------------------------------
// Launch
// ---------------------------------------------------------------------------
extern "C" void kernel_launch(void* const* d_in, const int* in_sizes, int n_in,
                              void* d_out, int out_size, void* d_ws, size_t ws_size,
                              hipStream_t stream) {
    const float* x = (const float*)d_in[0];
    const float* w = (const float*)d_in[1];
    const int B = in_sizes[0] / D_DIM;   // 65536
    const int N = in_sizes[1] / D_DIM;   // 4900
    float* out = (float*)d_out;

    float* x2 = (float*)d_ws;            // B floats
    float* w2 = x2 + B;                  // N floats

    som_rownorm_kernel<<<(B + 255) / 256, 256, 0, stream>>>(x, x2, B);
    som_rownorm_kernel<<<(N + 255) / 256, 256, 0, stream>>>(w, w2, N);

    const int NCHUNK  = 16;                       // N tiles per block
    const int n_tiles = (N + 15) / 16;            // 307
    const int m_tiles = (B + 15) / 16;            // 4096
    dim3 grid((n_tiles + NCHUNK - 1) / NCHUNK,    // 20
              (m_tiles + 7) / 8);                 // 512
    som_dist_kernel<<<grid, 256, 0, stream>>>(x, w, x2, w2, out, B, N, NCHUNK);
}